// MeshUnpool_15693810499887
// MI455X (gfx1250) — compile-verified
//
#include <hip/hip_runtime.h>
#include <stdint.h>

#define THREADS 256
#define WAVES_PER_BLOCK (THREADS / 32)
#define ROWS_PER_WAVE 4
#define SLOT_BYTES 512   // one pass of up to 128 floats

// ---------------------------------------------------------------------------
// Phase 1: init row maps: inv[row] = position in img (or -1), lw[row] = -1
// ---------------------------------------------------------------------------
__global__ __launch_bounds__(THREADS) void init_maps(int* __restrict__ inv,
                                                     int* __restrict__ lw, int N) {
    int r = blockIdx.x * THREADS + threadIdx.x;
    if (r < N) { inv[r] = -1; lw[r] = -1; }
}

__global__ __launch_bounds__(THREADS) void scatter_mask(const long long* __restrict__ mask_idx,
                                                        int* __restrict__ inv, int M) {
    int i = blockIdx.x * THREADS + threadIdx.x;
    if (i < M) inv[(int)mask_idx[i]] = i;
}

// ---------------------------------------------------------------------------
// Phase 2: step s corresponds to column j = K-1-s of `order` (reverse replay).
// fs[s] = order[0][j], ts[s] = order[1][j]; lw[row] = last step writing row.
// ---------------------------------------------------------------------------
__global__ __launch_bounds__(THREADS) void extract_steps(const long long* __restrict__ order,
                                                         int* __restrict__ fs,
                                                         int* __restrict__ ts,
                                                         int* __restrict__ lw, int K) {
    int s = blockIdx.x * THREADS + threadIdx.x;
    if (s >= K) return;
    int j = K - 1 - s;
    int f = (int)order[j];                 // order[0][j]
    int t = (int)order[(long long)K + j];  // order[1][j]
    fs[s] = f;
    ts[s] = t;
    atomicMax(&lw[t], s);
}

// ---------------------------------------------------------------------------
// Phase 3: anc[s] = last step s' < s with ts[s'] == fs[s] (parent link), -1 if
// none. root[s] = fs[s]. Backward scan is coalesced across lanes (lane i reads
// ts[s0+i-1-k]) and hits an 80KB L2-resident array.
// ---------------------------------------------------------------------------
__global__ __launch_bounds__(THREADS) void find_parent(const int* __restrict__ ts,
                                                       const int* __restrict__ fs,
                                                       int* __restrict__ anc,
                                                       int* __restrict__ root, int K) {
    int s = blockIdx.x * THREADS + threadIdx.x;
    if (s >= K) return;
    int f = fs[s];
    int p = -1;
    for (int k = s - 1; k >= 0; --k) {
        if (ts[k] == f) { p = k; break; }
    }
    anc[s] = p;
    root[s] = f;
}

// ---------------------------------------------------------------------------
// Phase 4: pointer doubling over the parent forest (anc[s] < s strictly, so no
// cycles). After convergence root[s] = original source row at step s.
// ---------------------------------------------------------------------------
__global__ __launch_bounds__(THREADS) void pointer_jump(const int* __restrict__ ai,
                                                        const int* __restrict__ ri,
                                                        int* __restrict__ ao,
                                                        int* __restrict__ ro, int K) {
    int s = blockIdx.x * THREADS + threadIdx.x;
    if (s >= K) return;
    int a = ai[s];
    if (a < 0) {
        ao[s] = -1;
        ro[s] = ri[s];
    } else {
        ao[s] = ai[a];
        ro[s] = ri[a];
    }
}

// ---------------------------------------------------------------------------
// Phase 5: bandwidth kernel. Each wave owns ROWS_PER_WAVE 512B LDS slots and
// moves ROWS_PER_WAVE rows per pass with the CDNA5 async global<->LDS DMA path:
//   [4x global_load_async_to_lds_b128] -> s_wait_asynccnt 0 ->
//   [4x global_store_async_from_lds_b128]
// Loads overlap loads and stores overlap stores; ASYNCcnt load/store
// completions are mutually unordered, so the single wait-0 between the phases
// is the only safe (and cheap) synchronization. Trailing stores are covered by
// S_ENDPGM's implicit wait-idle. Rows with no source are zero-filled with
// normal b128 stores. All per-row control flow is wave-uniform.
// ---------------------------------------------------------------------------
__global__ __launch_bounds__(THREADS) void gather_rows(const float* __restrict__ img,
                                                       const int* __restrict__ inv,
                                                       const int* __restrict__ lw,
                                                       const int* __restrict__ root,
                                                       float* __restrict__ out,
                                                       int N, int C) {
    __shared__ char ldsbuf[WAVES_PER_BLOCK * ROWS_PER_WAVE * SLOT_BYTES];
    const int wave = threadIdx.x >> 5;
    const int lane = threadIdx.x & 31;
    const long long waveId = (long long)blockIdx.x * WAVES_PER_BLOCK + wave;
    const long long row0 = waveId * ROWS_PER_WAVE;
    if (row0 >= N) return;

    // Workgroup-relative LDS byte offset: low 32 bits of the flat address.
    const unsigned ldsbase =
        (unsigned)(uintptr_t)(void*)&ldsbuf[wave * ROWS_PER_WAVE * SLOT_BYTES];

    const int passes = (C + 127) / 128;  // 1 for C == 128
    for (int p = 0; p < passes; ++p) {
        const int base = p * 128 + lane * 4;   // float offset handled by this lane
        const bool laneActive = (base < C);

        int jv[ROWS_PER_WAVE];

        // Resolve sources and issue the batch of async loads.
        for (int i = 0; i < ROWS_PER_WAVE; ++i) {
            const long long row = row0 + i;
            jv[i] = -1;
            if (row < N) {
                int s = lw[row];
                int src = (s >= 0) ? root[s] : (int)row;
                jv[i] = inv[src];
                if (jv[i] >= 0 && laneActive) {
                    unsigned long long ga =
                        (unsigned long long)(uintptr_t)(img + (long long)jv[i] * C + base);
                    unsigned lo = ldsbase + (unsigned)(i * SLOT_BYTES + lane * 16);
                    asm volatile("global_load_async_to_lds_b128 %0, %1, off"
                                 :: "v"(lo), "v"(ga) : "memory");
                }
            }
        }

        // All loads of this batch complete (in order among loads).
        asm volatile("s_wait_asynccnt 0" ::: "memory");

        // Issue the batch of async stores / zero fills.
        for (int i = 0; i < ROWS_PER_WAVE; ++i) {
            const long long row = row0 + i;
            if (row >= N) break;
            float* dstrow = out + row * (long long)C;
            if (jv[i] >= 0) {
                if (laneActive) {
                    unsigned long long gd =
                        (unsigned long long)(uintptr_t)(dstrow + base);
                    unsigned lo = ldsbase + (unsigned)(i * SLOT_BYTES + lane * 16);
                    asm volatile("global_store_async_from_lds_b128 %0, %1, off"
                                 :: "v"(gd), "v"(lo) : "memory");
                }
            } else if (laneActive) {
                float4 z = {0.0f, 0.0f, 0.0f, 0.0f};
                *reinterpret_cast<float4*>(dstrow + base) = z;
            }
        }

        // Before reusing the LDS slots in the next pass, drain the stores.
        if (p + 1 < passes) {
            asm volatile("s_wait_asynccnt 0" ::: "memory");
        }
    }
}

// ---------------------------------------------------------------------------
extern "C" void kernel_launch(void* const* d_in, const int* in_sizes, int n_in,
                              void* d_out, int out_size, void* d_ws, size_t ws_size,
                              hipStream_t stream) {
    const float* img          = (const float*)d_in[0];
    const long long* mask_idx = (const long long*)d_in[1];
    const long long* order    = (const long long*)d_in[2];
    // d_in[3] = num_vertices (scalar); N derived from out_size instead.

    const int M = in_sizes[1];      // 250000
    const int C = in_sizes[0] / M;  // 128
    const int K = in_sizes[2] / 2;  // 20000
    const int N = out_size / C;     // 500000

    // Workspace (ints): inv[N], lw[N], fs[K], ts[K], anc0[K], root0[K],
    // anc1[K], root1[K] -> (2N + 6K)*4 bytes ~= 4.5 MB.
    int* inv   = (int*)d_ws;
    int* lw    = inv + N;
    int* fs    = lw + N;
    int* ts    = fs + K;
    int* anc0  = ts + K;
    int* root0 = anc0 + K;
    int* anc1  = root0 + K;
    int* root1 = anc1 + K;

    dim3 blk(THREADS);
    int gN = (N + THREADS - 1) / THREADS;
    int gM = (M + THREADS - 1) / THREADS;
    int gK = (K + THREADS - 1) / THREADS;

    init_maps<<<gN, blk, 0, stream>>>(inv, lw, N);
    scatter_mask<<<gM, blk, 0, stream>>>(mask_idx, inv, M);
    extract_steps<<<gK, blk, 0, stream>>>(order, fs, ts, lw, K);
    find_parent<<<gK, blk, 0, stream>>>(ts, fs, anc0, root0, K);

    // ceil(log2(K)) doubling passes, rounded up to even so the final result
    // lands back in (anc0, root0).
    int iters = 1;
    while ((1LL << iters) < (long long)K) ++iters;
    if (iters & 1) ++iters;
    for (int it = 0; it < iters; ++it) {
        if ((it & 1) == 0)
            pointer_jump<<<gK, blk, 0, stream>>>(anc0, root0, anc1, root1, K);
        else
            pointer_jump<<<gK, blk, 0, stream>>>(anc1, root1, anc0, root0, K);
    }

    const long long totalWaves = ((long long)N + ROWS_PER_WAVE - 1) / ROWS_PER_WAVE;
    const long long gGather = (totalWaves + WAVES_PER_BLOCK - 1) / WAVES_PER_BLOCK;
    gather_rows<<<(unsigned)gGather, blk, 0, stream>>>(img, inv, lw, root0,
                                                       (float*)d_out, N, C);
}